// SSDTableBatchedEmbeddingBags_28887950033124
// MI455X (gfx1250) — compile-verified
//
#include <hip/hip_runtime.h>

// ---------------------------------------------------------------------------
// SSD table-batched embedding bag forward (SUM pooling) for MI455X / gfx1250.
//
// Memory-bound gather: ~860 MB of random 512-B row fetches -> use the CDNA5
// Tensor Data Mover in GATHER mode. One TENSOR_LOAD_TO_LDS with a gather D#
// moves 8 indexed embedding rows (4 KB) from global memory into LDS per
// instruction, tracked by TENSORcnt. One wave32 per bag, double-buffered
// LDS staging, ds_load_b128 accumulation, float4 output stores.
//
// NOTE: the TDM writes LDS behind the compiler's back (the descriptor holds
// integer LDS offsets, no pointers), so we must (a) escape the smem pointer
// and (b) make every tensorcnt wait a "memory" barrier, or LLVM proves smem
// is never written and deletes the ds_load accumulation entirely.
// ---------------------------------------------------------------------------

#define T_TABLES        4
#define BATCH           8192
#define DIM             128
#define ROWS_PER_TABLE  250000
#define TOTAL_ROWS      (T_TABLES * ROWS_PER_TABLE)   // 1,000,000 rows

#define WAVES_PER_BLOCK 4
#define CHUNK           8                    // rows per TDM gather op (32-bit idx mode max = 8)
#define ROW_BYTES       (DIM * 4)            // 512 B per embedding row
#define BUF_BYTES       (CHUNK * ROW_BYTES)  // 4 KB per staging buffer
#define WAVE_LDS        (2 * BUF_BYTES)      // 8 KB double buffer per wave

// float4 granularity: one buffer = CHUNK rows * 32 lanes = 256 float4
#define F4_PER_ROW      (DIM / 4)            // 32
#define F4_PER_BUF      (CHUNK * F4_PER_ROW) // 256
#define F4_PER_WAVE     (2 * F4_PER_BUF)     // 512

typedef unsigned int v4u __attribute__((ext_vector_type(4)));
typedef int          v8i __attribute__((ext_vector_type(8)));
typedef int          v4i __attribute__((ext_vector_type(4)));

__global__ __launch_bounds__(32 * WAVES_PER_BLOCK)
void tbe_fwd_tdm_kernel(const int* __restrict__ indices,
                        const int* __restrict__ offsets,
                        const float* __restrict__ weights,
                        float* __restrict__ out,
                        int num_bags)
{
    __shared__ float4 smem[WAVES_PER_BLOCK * F4_PER_WAVE];  // 32 KB, LDS offset 0

    // Escape the LDS buffer address so alias analysis must assume memory-
    // clobbering asm (our tensorcnt waits) can write it. Without this, LLVM
    // sees a never-stored non-escaping array and folds all loads to undef.
    {
        void* p = (void*)smem;
        asm volatile("" : "+v"(p) :: "memory");
    }

    const int lane = threadIdx.x & 31;
    const int wave = threadIdx.x >> 5;
    const int bag  = blockIdx.x * WAVES_PER_BLOCK + wave;
    if (bag >= num_bags) return;

    const int t = bag / BATCH;            // feature/table id
    const int b = bag - t * BATCH;        // batch id
    const int start = offsets[bag];
    const int end   = offsets[bag + 1];
    const int L     = end - start;

    // wave-uniform scalars for the descriptor
    const unsigned lds_wave_base =
        (unsigned)__builtin_amdgcn_readfirstlane(wave * WAVE_LDS);
    const int row_base = __builtin_amdgcn_readfirstlane(t * ROWS_PER_TABLE);
    const unsigned long long gaddr = (unsigned long long)(size_t)weights;
    const unsigned ga_lo = (unsigned)(gaddr & 0xffffffffULL);
    const unsigned ga_hi = (unsigned)((gaddr >> 32) & 0x1ffffffULL);

    // Issue one TDM gather-mode load of up to 8 rows into buffer (c & 1).
    auto issue_chunk = [&](int c) {
        const int remaining = L - c * CHUNK;
        int valid = remaining < CHUNK ? remaining : CHUNK;
        valid = __builtin_amdgcn_readfirstlane(valid);

        // lanes 0..valid-1 fetch this chunk's indices; linearize with table base
        int v = row_base;
        if (lane < valid) v = indices[start + c * CHUNK + lane] + row_base;

        const int r0 = __builtin_amdgcn_readlane(v, 0);
        const int r1 = __builtin_amdgcn_readlane(v, 1);
        const int r2 = __builtin_amdgcn_readlane(v, 2);
        const int r3 = __builtin_amdgcn_readlane(v, 3);
        const int r4 = __builtin_amdgcn_readlane(v, 4);
        const int r5 = __builtin_amdgcn_readlane(v, 5);
        const int r6 = __builtin_amdgcn_readlane(v, 6);
        const int r7 = __builtin_amdgcn_readlane(v, 7);

        const unsigned buf = lds_wave_base + (unsigned)((c & 1) * BUF_BYTES);

        // D# group 0: count=1 | gather_index_size=32b | gather_mode=1; lds/global addr; type=2
        v4u g0;
        g0.x = 0xC0000001u;
        g0.y = buf;
        g0.z = ga_lo;
        g0.w = ga_hi | (2u << 30);

        // D# group 1 (fields straddle dwords; see CDNA5 ISA):
        v8i g1;
        g1[0] = (int)(2u << 16);                              // data_size = 4 B, wg_mask = 0
        g1[1] = (int)((unsigned)DIM << 16);                   // tensor_dim0[15:0] = 128
        g1[2] = (int)(((unsigned)TOTAL_ROWS & 0xffffu) << 16);// dim0 hi=0 | tensor_dim1[15:0]
        g1[3] = (int)(((unsigned)DIM << 16) |
                      ((unsigned)TOTAL_ROWS >> 16));          // tensor_dim1 hi | tile_dim0=128
        g1[4] = valid;                                        // tile_dim1 = #valid gather indices
        g1[5] = DIM;                                          // tensor_dim0_stride = 128 elements
        g1[6] = 0;
        g1[7] = 0;

        // D# groups 2/3: eight 32-bit row indices
        v4i g2; g2[0] = r0; g2[1] = r1; g2[2] = r2; g2[3] = r3;
        v4i g3; g3[0] = r4; g3[1] = r5; g3[2] = r6; g3[3] = r7;

        // Fifth descriptor group (VADDR4 slot): unused per ISA -> zero-filled.
        v8i g4 = {0, 0, 0, 0, 0, 0, 0, 0};

        __builtin_amdgcn_tensor_load_to_lds(g0, g1, g2, g3, g4, 0);
    };

    const int nchunks = (L + CHUNK - 1) / CHUNK;

    if (nchunks > 0) issue_chunk(0);
    if (nchunks > 1) issue_chunk(1);

    float4 acc = make_float4(0.f, 0.f, 0.f, 0.f);

    for (int c = 0; c < nchunks; ++c) {
        // In-order per-wave TDM completion: oldest op done when count drops.
        // asm (not builtin) so the wait doubles as a may-write-LDS barrier:
        // the ds_loads below cannot be hoisted above it or folded away.
        if (c == nchunks - 1)
            asm volatile("s_wait_tensorcnt 0x0" ::: "memory");
        else
            asm volatile("s_wait_tensorcnt 0x1" ::: "memory");

        const int base = wave * F4_PER_WAVE + (c & 1) * F4_PER_BUF + lane;
        const int remaining = L - c * CHUNK;
        const int valid = remaining < CHUNK ? remaining : CHUNK;

#pragma unroll
        for (int j = 0; j < CHUNK; ++j) {
            if (j < valid) {
                const float4 r = smem[base + j * F4_PER_ROW];   // ds_load_b128
                acc.x += r.x; acc.y += r.y; acc.z += r.z; acc.w += r.w;
            }
        }

        // WAR guard: LDS reads must land before the TDM engine refills this buffer.
        asm volatile("s_wait_dscnt 0x0" ::: "memory");

        if (c + 2 < nchunks) issue_chunk(c + 2);
    }

    // out[b, t*D + 4*lane .. +4) — feature-major D layout
    float* dst = out + (size_t)b * (T_TABLES * DIM) + t * DIM + lane * 4;
    *reinterpret_cast<float4*>(dst) = acc;
}

extern "C" void kernel_launch(void* const* d_in, const int* in_sizes, int n_in,
                              void* d_out, int out_size, void* d_ws, size_t ws_size,
                              hipStream_t stream) {
    const int*   indices = (const int*)d_in[0];
    const int*   offsets = (const int*)d_in[1];
    const float* weights = (const float*)d_in[2];
    float*       out     = (float*)d_out;

    const int num_bags = in_sizes[1] - 1;   // T * B
    const int blocks = (num_bags + WAVES_PER_BLOCK - 1) / WAVES_PER_BLOCK;

    tbe_fwd_tdm_kernel<<<dim3(blocks), dim3(32 * WAVES_PER_BLOCK), 0, stream>>>(
        indices, offsets, weights, out, num_bags);
}